// GraphConv_3676492005525
// MI455X (gfx1250) — compile-verified
//
#include <hip/hip_runtime.h>

// CDNA5 / gfx1250 — wave32, WMMA 16x16x4 FP32 path.
typedef __attribute__((ext_vector_type(2))) float v2f;
typedef __attribute__((ext_vector_type(8))) float v8f;

static constexpr int FDIM = 128;          // F_IN == CHANNELS == 128
static constexpr int TILES_PER_BLK = 8;   // 8 waves per block, one 16-row tile each

// ---------------------------------------------------------------------------
// Kernel 1: XW = X @ W  (100000x128 @ 128x128, fp32, HBM-bound)
// One wave computes a 16x128 output strip via 32 K-steps of
// v_wmma_f32_16x16x4_f32 into 8 x v8f accumulators. W staged in LDS (64 KB).
// ---------------------------------------------------------------------------
__global__ __launch_bounds__(256)
void gemm_xw_wmma(const float* __restrict__ X, const float* __restrict__ W,
                  float* __restrict__ XW, int n_nodes)
{
    __shared__ float Wl[FDIM * FDIM];     // 64 KB of the 320 KB WGP LDS

    const int tid = threadIdx.x;
    {   // cooperative, fully coalesced float4 load of W into LDS
        const float4* Wg4 = (const float4*)W;
        float4*       Wl4 = (float4*)Wl;
        #pragma unroll
        for (int i = 0; i < (FDIM * FDIM / 4) / 256; ++i)
            Wl4[tid + i * 256] = Wg4[tid + i * 256];
    }
    __syncthreads();

    const int wave      = tid >> 5;
    const int lane      = tid & 31;
    const int rowTile   = blockIdx.x * TILES_PER_BLK + wave;
    const int nRowTiles = n_nodes >> 4;   // 100000 / 16 = 6250 exactly
    if (rowTile >= nRowTiles) return;

    const int m0    = rowTile << 4;
    const int lrow  = lane & 15;          // M (for A) / N (for B) within tile
    const int khalf = (lane >> 4) << 1;   // lanes 16-31 hold K+2,K+3

    v8f acc[8];
    #pragma unroll
    for (int n = 0; n < 8; ++n) acc[n] = (v8f)0.0f;

    // A fragment source: row m0+lrow, contiguous pair of K values -> b64 loads
    const float* arow = X + (size_t)(m0 + lrow) * FDIM + khalf;

    #pragma unroll 8
    for (int k0 = 0; k0 < FDIM; k0 += 4) {
        v2f a;
        a.x = arow[k0 + 0];
        a.y = arow[k0 + 1];
        const float* bcolp = Wl + (k0 + khalf) * FDIM + lrow;
        #pragma unroll
        for (int n = 0; n < 8; ++n) {     // 8 column tiles of 16
            v2f b;
            b.x = bcolp[n * 16];          // W[k0+khalf  ][n*16+lrow]
            b.y = bcolp[n * 16 + FDIM];   // W[k0+khalf+1][n*16+lrow]
            acc[n] = __builtin_amdgcn_wmma_f32_16x16x4_f32(
                false, a, false, b, (short)0, acc[n], false, false);
        }
    }

    // C/D layout: VGPR r, lanes 0-15 -> row m0+r, lanes 16-31 -> row m0+8+r
    const int rbase = (lane >> 4) << 3;
    #pragma unroll
    for (int n = 0; n < 8; ++n) {
        #pragma unroll
        for (int r = 0; r < 8; ++r) {
            XW[(size_t)(m0 + rbase + r) * FDIM + n * 16 + lrow] = acc[n][r];
        }
    }
}

// ---------------------------------------------------------------------------
// Kernel 2: out = relu(A_tilde @ XW + bias), COO with sorted rows.
// One wave per destination node: uniform binary search for the node's edge
// range, then per-lane float4 (4 channels) register accumulation. Gathers of
// XW rows hit the 192 MB L2 (XW is 51.2 MB). No atomics needed.
// ---------------------------------------------------------------------------
__global__ __launch_bounds__(256)
void spmm_bias_relu(const float* __restrict__ XW, const float* __restrict__ bias,
                    const float* __restrict__ eval, const int* __restrict__ erow,
                    const int* __restrict__ ecol, float* __restrict__ out,
                    int n_nodes, int n_edges)
{
    const int wave = threadIdx.x >> 5;
    const int lane = threadIdx.x & 31;
    const int node = blockIdx.x * 8 + wave;
    if (node >= n_nodes) return;

    // lower_bound(node), then upper_bound(node) over sorted erow (wave-uniform)
    int lo = 0, hi = n_edges;
    while (lo < hi) {
        int mid = (lo + hi) >> 1;
        if (erow[mid] < node) lo = mid + 1; else hi = mid;
    }
    const int start = lo;
    hi = n_edges;
    while (lo < hi) {
        int mid = (lo + hi) >> 1;
        if (erow[mid] <= node) lo = mid + 1; else hi = mid;
    }
    const int end = lo;

    float4 acc = make_float4(0.f, 0.f, 0.f, 0.f);
    for (int e = start; e < end; ++e) {
        const int   c = ecol[e];          // wave-uniform
        const float v = eval[e];          // wave-uniform
        const float4 s = ((const float4*)(XW + (size_t)c * FDIM))[lane]; // b128, coalesced
        acc.x += s.x * v; acc.y += s.y * v; acc.z += s.z * v; acc.w += s.w * v;
    }

    const float4 b = ((const float4*)bias)[lane];
    acc.x = fmaxf(acc.x + b.x, 0.f);
    acc.y = fmaxf(acc.y + b.y, 0.f);
    acc.z = fmaxf(acc.z + b.z, 0.f);
    acc.w = fmaxf(acc.w + b.w, 0.f);
    ((float4*)(out + (size_t)node * FDIM))[lane] = acc;
}

// ---------------------------------------------------------------------------
// Host launcher. Inputs: x, kernel, bias, edge_val, edge_row, edge_col.
// d_ws holds XW (n_nodes * 128 floats = 51.2 MB), fully written before read.
// ---------------------------------------------------------------------------
extern "C" void kernel_launch(void* const* d_in, const int* in_sizes, int n_in,
                              void* d_out, int out_size, void* d_ws, size_t ws_size,
                              hipStream_t stream) {
    const float* x    = (const float*)d_in[0];
    const float* w    = (const float*)d_in[1];
    const float* bias = (const float*)d_in[2];
    const float* ev   = (const float*)d_in[3];
    const int*   er   = (const int*)d_in[4];
    const int*   ec   = (const int*)d_in[5];
    float* out = (float*)d_out;
    float* xw  = (float*)d_ws;

    const int n_nodes = in_sizes[0] / FDIM;
    const int n_edges = in_sizes[3];

    const int nRowTiles = (n_nodes + 15) / 16;
    const int grid1 = (nRowTiles + TILES_PER_BLK - 1) / TILES_PER_BLK;
    hipLaunchKernelGGL(gemm_xw_wmma, dim3(grid1), dim3(256), 0, stream,
                       x, w, xw, n_nodes);

    const int grid2 = (n_nodes + 7) / 8;
    hipLaunchKernelGGL(spmm_bias_relu, dim3(grid2), dim3(256), 0, stream,
                       xw, bias, ev, er, ec, out, n_nodes, n_edges);
}